// CrossAttention_83167746719748
// MI455X (gfx1250) — compile-verified
//
#include <hip/hip_runtime.h>
#include <hip/hip_bf16.h>

#define DIM 1024
#define HEAD_DIM 64
#define NUM_HEADS 16
#define BATCH 2
#define SEQ_N 1024
#define SEQ_M 2048
#define QK_SCALE 0.125f   // 64^-0.5

typedef __attribute__((ext_vector_type(2))) float v2f;
typedef __attribute__((ext_vector_type(8))) float v8f;

__device__ __forceinline__ v8f wmma4(v2f a, v2f b, v8f c) {
    // D = A(16x4 f32) * B(4x16 f32) + C(16x16 f32), full fp32 precision
    return __builtin_amdgcn_wmma_f32_16x16x4_f32(
        /*neg_a=*/false, a, /*neg_b=*/false, b,
        /*c_mod=*/(short)0, c, /*reuse_a=*/false, /*reuse_b=*/false);
}

__device__ __forceinline__ v2f ld2(const float* p) { return *(const v2f*)p; }

// ---------------------------------------------------------------------------
// C[Mr x Nc] = A[Mr x K] @ W[Nc x K]^T (+ bias[Nc]).  A, W row-major (K contig).
// Block = 128 threads (4 waves). Each wave computes a 16x64 strip of C.
// ---------------------------------------------------------------------------
template<bool BIAS>
__global__ __launch_bounds__(128)
void gemm_xwt(const float* __restrict__ A, const float* __restrict__ W,
              const float* __restrict__ bias, float* __restrict__ C,
              int K, int Nc) {
    const int lane = threadIdx.x & 31;
    const int wave = threadIdx.x >> 5;
    const int lm = lane & 15;          // M index (A) / N index (B)
    const int lh = lane >> 4;          // lane-half selects K pair
    const int m0 = blockIdx.y * 64 + wave * 16;
    const int nb = blockIdx.x * 64;

    const float* arow = A + (size_t)(m0 + lm) * K + 2 * lh;
    const float* wrow = W + (size_t)(nb + lm) * K + 2 * lh;

    v8f acc[4] = {v8f{}, v8f{}, v8f{}, v8f{}};

    #pragma unroll 4
    for (int k = 0; k < K; k += 4) {
        v2f a = ld2(arow + k);
        acc[0] = wmma4(a, ld2(wrow + (size_t)0  * K + k), acc[0]);
        acc[1] = wmma4(a, ld2(wrow + (size_t)16 * K + k), acc[1]);
        acc[2] = wmma4(a, ld2(wrow + (size_t)32 * K + k), acc[2]);
        acc[3] = wmma4(a, ld2(wrow + (size_t)48 * K + k), acc[3]);
    }

    #pragma unroll
    for (int t = 0; t < 4; ++t) {
        const int col = nb + t * 16 + lm;
        const float bv = BIAS ? bias[col] : 0.0f;
        #pragma unroll
        for (int j = 0; j < 8; ++j) {
            const int row = m0 + j + 8 * lh;       // C frag: row = vgpr + 8*half
            C[(size_t)row * Nc + col] = acc[t][j] + bv;
        }
    }
}

// ---------------------------------------------------------------------------
// Flash attention: one wave per (b, h, 16-row Q tile), streaming 16 keys/step.
// Q: [B, N, DIM] (head-concat),  KV: [B, M, 2, H, D],  mask: [B, N, M] u8.
// ---------------------------------------------------------------------------
__global__ __launch_bounds__(32)
void attn_flash(const float* __restrict__ Q, const float* __restrict__ KV,
                const unsigned char* __restrict__ mask, float* __restrict__ O) {
    __shared__ float Stile[16 * 18];   // stride 18 keeps float2 reads 8B aligned
    __shared__ float rm[16], rl[16], rc[16];

    const int lane = threadIdx.x & 31;
    const int lm = lane & 15;
    const int lh = lane >> 4;
    const int b  = blockIdx.z;
    const int h  = blockIdx.y;
    const int n0 = blockIdx.x * 16;

    // Preload Q A-frags (16 k-steps of 4 over HEAD_DIM=64) into registers.
    const float* qbase = Q + (size_t)(b * SEQ_N + n0 + lm) * DIM + h * HEAD_DIM + 2 * lh;
    v2f qf[16];
    #pragma unroll
    for (int kk = 0; kk < 16; ++kk) qf[kk] = ld2(qbase + 4 * kk);

    const float* kbase = KV + (size_t)b * SEQ_M * 2 * DIM + h * HEAD_DIM; // + m*2*DIM + d
    const float* vbase = kbase + DIM;                                     // s=1 slot
    const unsigned char* mbase = mask + (size_t)(b * SEQ_N + n0) * SEQ_M;

    if (lane < 16) { rm[lane] = -__builtin_inff(); rl[lane] = 0.0f; }
    __syncthreads();

    v8f o0 = {}, o1 = {}, o2 = {}, o3 = {};

    for (int m0 = 0; m0 < SEQ_M; m0 += 16) {
        // ---- S = Q @ K_chunk^T  (16 WMMAs) ----
        v8f s = {};
        const float* kf = kbase + (size_t)(m0 + lm) * (2 * DIM) + 2 * lh;
        #pragma unroll
        for (int kk = 0; kk < 16; ++kk)
            s = wmma4(qf[kk], ld2(kf + 4 * kk), s);

        #pragma unroll
        for (int j = 0; j < 8; ++j)
            Stile[(j + 8 * lh) * 18 + lm] = s[j] * QK_SCALE;
        __syncthreads();

        // ---- online softmax, one row per lane (lanes 0..15) ----
        if (lane < 16) {
            const unsigned char* mr = mbase + (size_t)lane * SEQ_M + m0;
            float mold = rm[lane];
            float cmax = -__builtin_inff();
            float sv[16];
            #pragma unroll
            for (int c = 0; c < 16; ++c) {
                float v = mr[c] ? Stile[lane * 18 + c] : -__builtin_inff();
                sv[c] = v;
                cmax = fmaxf(cmax, v);
            }
            float mnew = fmaxf(mold, cmax);
            float corr, rs = 0.0f;
            if (mnew == -__builtin_inff()) {       // fully-masked so far
                corr = 1.0f;
                #pragma unroll
                for (int c = 0; c < 16; ++c) Stile[lane * 18 + c] = 0.0f;
            } else {
                corr = __expf(mold - mnew);        // mold=-inf -> 0
                #pragma unroll
                for (int c = 0; c < 16; ++c) {
                    float p = __expf(sv[c] - mnew); // masked -inf -> 0
                    Stile[lane * 18 + c] = p;
                    rs += p;
                }
            }
            rm[lane] = mnew;
            rl[lane] = rl[lane] * corr + rs;
            rc[lane] = corr;
        }
        __syncthreads();

        // ---- rescale O accumulator rows ----
        float cj[8];
        #pragma unroll
        for (int j = 0; j < 8; ++j) cj[j] = rc[j + 8 * lh];
        #pragma unroll
        for (int j = 0; j < 8; ++j) {
            o0[j] *= cj[j]; o1[j] *= cj[j]; o2[j] *= cj[j]; o3[j] *= cj[j];
        }

        // ---- O += P @ V_chunk  (16 WMMAs) ----
        #pragma unroll
        for (int kk = 0; kk < 4; ++kk) {
            v2f pa = ld2(&Stile[lm * 18 + kk * 4 + 2 * lh]);   // P in A-frag layout
            const float* vr0 = vbase + (size_t)(m0 + kk * 4 + 2 * lh) * (2 * DIM);
            const float* vr1 = vr0 + 2 * DIM;
            v2f vb;
            vb.x = vr0[ 0 + lm]; vb.y = vr1[ 0 + lm]; o0 = wmma4(pa, vb, o0);
            vb.x = vr0[16 + lm]; vb.y = vr1[16 + lm]; o1 = wmma4(pa, vb, o1);
            vb.x = vr0[32 + lm]; vb.y = vr1[32 + lm]; o2 = wmma4(pa, vb, o2);
            vb.x = vr0[48 + lm]; vb.y = vr1[48 + lm]; o3 = wmma4(pa, vb, o3);
        }
        __syncthreads();
    }

    // ---- finalize: divide by row sums, store head-concat O ----
    float inv[8];
    #pragma unroll
    for (int j = 0; j < 8; ++j) {
        float l = rl[j + 8 * lh];
        inv[j] = (l > 0.0f) ? 1.0f / l : 0.0f;
    }
    float* obase = O + (size_t)(b * SEQ_N + n0) * DIM + h * HEAD_DIM;
    #pragma unroll
    for (int j = 0; j < 8; ++j) {
        float* orow = obase + (size_t)(j + 8 * lh) * DIM;
        orow[ 0 + lm] = o0[j] * inv[j];
        orow[16 + lm] = o1[j] * inv[j];
        orow[32 + lm] = o2[j] * inv[j];
        orow[48 + lm] = o3[j] * inv[j];
    }
}

// ---------------------------------------------------------------------------
extern "C" void kernel_launch(void* const* d_in, const int* in_sizes, int n_in,
                              void* d_out, int out_size, void* d_ws, size_t ws_size,
                              hipStream_t stream) {
    const float* x    = (const float*)d_in[0];   // [B, N, DIM]
    const float* ctx  = (const float*)d_in[1];   // [B, M, DIM]
    const unsigned char* mask = (const unsigned char*)d_in[2]; // [B, N, M] bool
    const float* Wq   = (const float*)d_in[3];   // [DIM, DIM]
    const float* Wkv  = (const float*)d_in[4];   // [2*DIM, DIM]
    const float* bkv  = (const float*)d_in[5];   // [2*DIM]
    const float* Wo   = (const float*)d_in[6];   // [DIM, DIM]
    float* out = (float*)d_out;                  // [B, N, DIM]

    // Workspace: Q (8MB) + KV (32MB) + O (8MB) = 48MB
    float* Qb  = (float*)d_ws;
    float* KVb = Qb  + (size_t)BATCH * SEQ_N * DIM;
    float* Ob  = KVb + (size_t)BATCH * SEQ_M * 2 * DIM;

    dim3 blk(128);
    // Q = x @ Wq^T
    gemm_xwt<false><<<dim3(DIM / 64, BATCH * SEQ_N / 64), blk, 0, stream>>>(
        x, Wq, nullptr, Qb, DIM, DIM);
    // KV = ctx @ Wkv^T + b_kv
    gemm_xwt<true><<<dim3(2 * DIM / 64, BATCH * SEQ_M / 64), blk, 0, stream>>>(
        ctx, Wkv, bkv, KVb, DIM, 2 * DIM);
    // masked softmax attention (flash-style)
    attn_flash<<<dim3(SEQ_N / 16, NUM_HEADS, BATCH), 32, 0, stream>>>(
        Qb, KVb, mask, Ob);
    // out = O @ Wo^T
    gemm_xwt<false><<<dim3(DIM / 64, BATCH * SEQ_N / 64), blk, 0, stream>>>(
        Ob, Wo, nullptr, out, DIM, DIM);
}